// PrecisionGatedMoE_90804198572580
// MI455X (gfx1250) — compile-verified
//
#include <hip/hip_runtime.h>
#include <hip/hip_bf16.h>

typedef float v2f __attribute__((ext_vector_type(2)));
typedef float v4f __attribute__((ext_vector_type(4)));
typedef float v8f __attribute__((ext_vector_type(8)));

// Problem shapes (fixed by the reference harness)
constexpr int H = 4096;   // hidden dim
constexpr int E = 8;      // experts

constexpr int NWAVES     = 8;   // waves per block; each covers H/NWAVES of K
constexpr int KCHUNK     = 64;  // K staged into LDS per iteration
constexpr int LDS_STRIDE = 68;  // 64 + 4 pad -> conflict-free WMMA A reads
constexpr int KSPAN      = H / NWAVES;  // 512

// ---------------- Kernel 1: gate GEMM (WMMA f32 16x16x4) + top-2 ----------------
// Block = 16 rows. Each wave accumulates a 16x16 score tile over its private
// K-span; partials reduced through LDS. Accumulator columns 8..15 collect a
// duplicate of column 0 (never read) so no B-masking is needed.
__global__ __launch_bounds__(NWAVES * 32) void moe_gate_wmma(
    const float* __restrict__ hidden,   // [B][H]
    const float* __restrict__ gw,       // [H][E]
    const float* __restrict__ gb,       // [E]
    float*       __restrict__ ws)       // [B][4] : q0, q1, bitcast(i0), bitcast(i1)
{
    __shared__ float ldsA[NWAVES][16][LDS_STRIDE];
    __shared__ float ldsS[NWAVES][16][16];

    const int tid  = threadIdx.x;
    const int wave = tid >> 5;
    const int lane = tid & 31;
    const int row0 = blockIdx.x * 16;

    const int kw0 = wave * KSPAN;

    const int m  = lane & 15;            // row-in-tile for A; N column for B/C
    const int hh = lane >> 4;            // half-wave select
    const int kh = hh * 2;               // K sub-offset within a 16x16x4 frag
    const int nn = (m < 8) ? m : 0;      // safe in-bounds gate_w column

    // Per-lane base pointers: everything below is base + compile-time immediate.
    const float* hp  = hidden + (size_t)(row0 + hh) * H + kw0 + m * 4;  // A source
    float*       lap = &ldsA[wave][hh][m * 4];                          // A dest
    const float* lar = &ldsA[wave][m][kh];                              // A frag reads
    const float* gwc = gw + (size_t)kw0 * E + (size_t)kh * E + nn;      // B source

    v8f acc = {0.f, 0.f, 0.f, 0.f, 0.f, 0.f, 0.f, 0.f};

    for (int kc = 0; kc < KSPAN; kc += KCHUNK) {
        // Stage A: 16 x KCHUNK fp32; 256B-contiguous per half-wave (b128, imm offsets)
        #pragma unroll
        for (int rp = 0; rp < 8; ++rp) {
            v4f v = *(const v4f*)(hp + rp * 2 * H);
            *(v4f*)(lap + rp * 2 * LDS_STRIDE) = v;
        }

        // 16 WMMAs over the chunk; A from LDS (b64, imm offset), B from L2-resident gw
        #pragma unroll
        for (int kk = 0; kk < KCHUNK; kk += 4) {
            v2f a = *(const v2f*)(lar + kk);       // 8B aligned: kk+kh even
            v2f b;
            b.x = gwc[(kk + 0) * E];
            b.y = gwc[(kk + 1) * E];
            acc = __builtin_amdgcn_wmma_f32_16x16x4_f32(
                false, a, false, b, (short)0, acc, false, false);
        }

        hp  += KCHUNK;
        gwc += KCHUNK * E;
    }

    // C/D layout: lanes 0-15 hold rows 0..7 (acc[0..7]) at N=lane; lanes 16-31 rows 8..15
    const int rbase = hh * 8;
    #pragma unroll
    for (int v = 0; v < 8; ++v)
        ldsS[wave][rbase + v][m] = acc[v];
    __syncthreads();

    if (tid < 16) {
        const int r = tid;
        float s[E];
        #pragma unroll
        for (int e = 0; e < E; ++e) {
            float acc_e = gb[e];
            #pragma unroll
            for (int w = 0; w < NWAVES; ++w) acc_e += ldsS[w][r][e];
            s[e] = acc_e;
        }
        // top-2; ties keep the lower index (matches jax.lax.top_k)
        float b0 = s[0]; int i0 = 0;
        float b1 = -3.402823466e38f; int i1 = 0;
        #pragma unroll
        for (int e = 1; e < E; ++e) {
            if (s[e] > b0)      { b1 = b0; i1 = i0; b0 = s[e]; i0 = e; }
            else if (s[e] > b1) { b1 = s[e]; i1 = e; }
        }
        // renormalized top-2 softmax == sigmoid of the score gap
        const float q0 = 1.0f / (1.0f + __expf(b1 - b0));
        v4f o;
        o.x = q0;
        o.y = 1.0f - q0;
        o.z = __int_as_float(i0);
        o.w = __int_as_float(i1);
        *(v4f*)(ws + (size_t)(row0 + r) * 4) = o;
    }
}

// ---------------- Kernel 2: streaming weighted combine (bandwidth phase) ----------------
// One block per row; all data touch-once -> nontemporal b128 with imm offsets.
__global__ __launch_bounds__(256) void moe_combine(
    const float* __restrict__ eo,       // [E][B][H]
    const float* __restrict__ ws,       // [B][4]
    float*       __restrict__ out,      // [B][H]
    long long BH)
{
    const int row = blockIdx.x;
    const v4f p = *(const v4f*)(ws + (size_t)row * 4);
    const float q0 = p.x, q1 = p.y;
    const long long e0 = (long long)__float_as_int(p.z);
    const long long e1 = (long long)__float_as_int(p.w);
    const float* s0 = eo + e0 * BH + (long long)row * H + threadIdx.x * 4;
    const float* s1 = eo + e1 * BH + (long long)row * H + threadIdx.x * 4;
    float*       o  = out + (size_t)row * H + threadIdx.x * 4;

    #pragma unroll
    for (int it = 0; it < H / (256 * 4); ++it) {
        v4f a = __builtin_nontemporal_load((const v4f*)(s0 + it * 1024));
        v4f b = __builtin_nontemporal_load((const v4f*)(s1 + it * 1024));
        v4f r = q0 * a + q1 * b;
        __builtin_nontemporal_store(r, (v4f*)(o + it * 1024));
    }
}

extern "C" void kernel_launch(void* const* d_in, const int* in_sizes, int n_in,
                              void* d_out, int out_size, void* d_ws, size_t ws_size,
                              hipStream_t stream) {
    const float* hidden = (const float*)d_in[0];   // [B][H]
    const float* eo     = (const float*)d_in[1];   // [E][B][H]
    const float* gw     = (const float*)d_in[2];   // [H][E]
    const float* gb     = (const float*)d_in[3];   // [E]
    float* out = (float*)d_out;
    float* ws  = (float*)d_ws;

    const long long B = (long long)in_sizes[0] / H;   // 8192

    moe_gate_wmma<<<(int)(B / 16), NWAVES * 32, 0, stream>>>(hidden, gw, gb, ws);
    moe_combine<<<(int)B, 256, 0, stream>>>(eo, ws, out, (long long)B * H);
}